// RightPooling_70927089926350
// MI455X (gfx1250) — compile-verified
//
#include <hip/hip_runtime.h>

// cummax along T for a [B, T, H, C] = [16, 512, 64, 128] fp32 tensor.
// Pure streaming problem: 512 MB total traffic -> ~22 us floor at 23.3 TB/s.
// One thread owns one float4 column (4 contiguous channels) and scans T
// sequentially. Unrolled loads give in-wave MLP; global_prefetch_b8 of the
// next tile hides HBM latency behind the current tile; NT stores keep the
// output stream from thrashing L2 (which caches the prefetched input).

typedef float v4f __attribute__((ext_vector_type(4)));

#define T_LEN       512
#define ROW_STRIDE  8192            // H*C floats between consecutive t
#define COL4_PER_B  (ROW_STRIDE/4)  // 2048 float4 columns per batch
#define UNROLL      16

__global__ __launch_bounds__(64) void cummax_t_kernel(
    const float* __restrict__ in, float* __restrict__ out, int ncol4)
{
    int tid = blockIdx.x * blockDim.x + threadIdx.x;
    if (tid >= ncol4) return;

    const int b    = tid >> 11;                 // tid / COL4_PER_B
    const int c4   = tid & (COL4_PER_B - 1);
    const int base = b * (T_LEN * ROW_STRIDE) + c4 * 4;

    const float* __restrict__ pin  = in  + base;
    float* __restrict__       pout = out + base;

    v4f m;
    m.x = -__builtin_inff();
    m.y = m.x; m.z = m.x; m.w = m.x;

    for (int t0 = 0; t0 < T_LEN; t0 += UNROLL) {
        // Prefetch the NEXT tile of time-slices into L2 (gfx1250
        // global_prefetch_b8) so the demand loads of the following
        // iteration hit cache instead of paying full HBM latency.
        if (t0 + UNROLL < T_LEN) {
#pragma unroll
            for (int u = 0; u < UNROLL; ++u)
                __builtin_prefetch(pin + (t0 + UNROLL + u) * ROW_STRIDE, 0, 0);
        }

        // Issue all UNROLL independent 128-bit loads up front (MLP).
        v4f v[UNROLL];
#pragma unroll
        for (int u = 0; u < UNROLL; ++u)
            v[u] = *(const v4f*)(pin + (t0 + u) * ROW_STRIDE);

        // Dependent running-max chain + streaming NT stores.
#pragma unroll
        for (int u = 0; u < UNROLL; ++u) {
            m.x = fmaxf(m.x, v[u].x);
            m.y = fmaxf(m.y, v[u].y);
            m.z = fmaxf(m.z, v[u].z);
            m.w = fmaxf(m.w, v[u].w);
            __builtin_nontemporal_store(m, (v4f*)(pout + (t0 + u) * ROW_STRIDE));
        }
    }
}

extern "C" void kernel_launch(void* const* d_in, const int* in_sizes, int n_in,
                              void* d_out, int out_size, void* d_ws, size_t ws_size,
                              hipStream_t stream) {
    (void)n_in; (void)d_ws; (void)ws_size; (void)out_size;
    const float* in  = (const float*)d_in[0];
    float*       out = (float*)d_out;

    // total elements = B * T * H * C; float4 columns = total / (T * 4)
    const int total = in_sizes[0];
    const int ncol4 = total / (T_LEN * 4);

    const int block = 64;                       // 2 wave32s per workgroup
    const int grid  = (ncol4 + block - 1) / block;
    cummax_t_kernel<<<grid, block, 0, stream>>>(in, out, ncol4);
}